// HGCNConv_35880156791578
// MI455X (gfx1250) — compile-verified
//
#include <hip/hip_runtime.h>
#include <stdint.h>

// HGCNConv: out = LeakyReLU_{0.5}( A * (A^T * X) ) with A sparse (COO).
// Memory-bound scatter/gather SpMM; working set fits MI455X 192MB L2.
// CDNA5 paths used: global_load_async_to_lds_b128 (double-buffered index
// staging, ASYNCcnt-managed) + native global_atomic_add_f32 L2 atomics.

#define BLOCK 256
#define TILE  1024          // nnz per LDS tile (NNZ=3.2M is a multiple)
#define LOG2D 7             // D = 128

// Low 32 bits of a generic pointer into LDS == LDS byte offset
// (aperture mapping).
__device__ __forceinline__ uint32_t lds_off(const void* p) {
    return (uint32_t)(uintptr_t)p;
}

// Async DMA: 16 bytes per lane, global -> LDS. Tracked by ASYNCcnt.
__device__ __forceinline__ void async_g2l_b128(uint32_t lo, const void* g) {
    asm volatile("global_load_async_to_lds_b128 %0, %1, off"
                 :: "v"(lo), "v"((unsigned long long)(uintptr_t)g)
                 : "memory");
}

__global__ void __launch_bounds__(BLOCK)
spmm_scatter_kernel(const float* __restrict__ vals,
                    const int*   __restrict__ gidx,   // gather row per nnz
                    const int*   __restrict__ sidx,   // scatter row per nnz
                    const float* __restrict__ src,    // [*, 128] gathered
                    float*       __restrict__ dst,    // [*, 128] atomic accum
                    int nnz)
{
    __shared__ float s_vals[2][TILE];
    __shared__ int   s_g[2][TILE];
    __shared__ int   s_s[2][TILE];

    const int tid  = threadIdx.x;
    const int wave = tid >> 5;
    const int lane = tid & 31;
    const int numTiles = (nnz + TILE - 1) / TILE;

    // Stage one tile of (val, gidx, sidx) into LDS buffer b.
    // Returns true if the async path was used (3 ASYNCcnt ops per wave).
    auto copy_tile = [&](int t, int b) -> bool {
        long base = (long)t * TILE;
        int  cnt  = min(TILE, nnz - (int)base);
        if (cnt == TILE) {
            int e = tid * 4;  // 4 dwords per thread per array = 4KB/array
            async_g2l_b128(lds_off(&s_vals[b][e]), vals + base + e);
            async_g2l_b128(lds_off(&s_g[b][e]),    gidx + base + e);
            async_g2l_b128(lds_off(&s_s[b][e]),    sidx + base + e);
            return true;
        }
        for (int j = tid; j < cnt; j += BLOCK) {   // rare partial tail tile
            s_vals[b][j] = vals[base + j];
            s_g[b][j]    = gidx[base + j];
            s_s[b][j]    = sidx[base + j];
        }
        return false;
    };

    int t   = blockIdx.x;
    int cur = 0;
    if (t < numTiles) copy_tile(t, cur);

    for (; t < numTiles; t += gridDim.x) {
        int  nt = t + gridDim.x;
        bool pf = false;
        if (nt < numTiles) pf = copy_tile(nt, cur ^ 1);   // prefetch next tile

        // Async ops complete in order: allowing 3 outstanding leaves only the
        // freshly issued prefetch in flight; current tile is guaranteed done.
        if (pf) asm volatile("s_wait_asynccnt 0x3" ::: "memory");
        else    asm volatile("s_wait_asynccnt 0x0" ::: "memory");
        __syncthreads();   // cross-wave visibility of LDS tile

        int cnt = min(TILE, nnz - t * TILE);
        for (int j = wave; j < cnt; j += BLOCK / 32) {   // one nnz per wave
            float v = s_vals[cur][j];                    // LDS broadcast reads
            int   r = s_g[cur][j];
            int   c = s_s[cur][j];
            const float4 e = *(const float4*)(src + ((size_t)r << LOG2D) + (lane << 2));
            float* p = dst + ((size_t)c << LOG2D) + (lane << 2);
            unsafeAtomicAdd(p + 0, v * e.x);   // global_atomic_add_f32 (L2)
            unsafeAtomicAdd(p + 1, v * e.y);
            unsafeAtomicAdd(p + 2, v * e.z);
            unsafeAtomicAdd(p + 3, v * e.w);
        }
        __syncthreads();   // done reading buf[cur] before it is refilled
        cur ^= 1;
    }
}

__global__ void zero_f32_kernel(float4* __restrict__ p, long n4) {
    long i      = (long)blockIdx.x * blockDim.x + threadIdx.x;
    long stride = (long)gridDim.x * blockDim.x;
    float4 z = make_float4(0.f, 0.f, 0.f, 0.f);
    for (; i < n4; i += stride) p[i] = z;
}

// H (#hyperedges) is only available as a device scalar -> read it on device.
__global__ void zero_hyper_kernel(float4* __restrict__ p, const int* __restrict__ pH) {
    long n4     = ((long)(*pH) << LOG2D) >> 2;
    long i      = (long)blockIdx.x * blockDim.x + threadIdx.x;
    long stride = (long)gridDim.x * blockDim.x;
    float4 z = make_float4(0.f, 0.f, 0.f, 0.f);
    for (; i < n4; i += stride) p[i] = z;
}

__global__ void leaky_kernel(float4* __restrict__ p, long n4) {
    long i      = (long)blockIdx.x * blockDim.x + threadIdx.x;
    long stride = (long)gridDim.x * blockDim.x;
    for (; i < n4; i += stride) {
        float4 x = p[i];
        x.x = (x.x >= 0.f) ? x.x : 0.5f * x.x;
        x.y = (x.y >= 0.f) ? x.y : 0.5f * x.y;
        x.z = (x.z >= 0.f) ? x.z : 0.5f * x.z;
        x.w = (x.w >= 0.f) ? x.w : 0.5f * x.w;
        p[i] = x;
    }
}

extern "C" void kernel_launch(void* const* d_in, const int* in_sizes, int n_in,
                              void* d_out, int out_size, void* d_ws, size_t ws_size,
                              hipStream_t stream) {
    const float* vals = (const float*)d_in[0];   // [NNZ] f32
    const float* embs = (const float*)d_in[1];   // [N,128] f32
    const int*   rows = (const int*)d_in[2];     // [NNZ] i32
    const int*   cols = (const int*)d_in[3];     // [NNZ] i32
    const int*   dH   = (const int*)d_in[5];     // device scalar: n_hyperedges
    float* out   = (float*)d_out;                // [N,128] f32
    float* hyper = (float*)d_ws;                 // [H,128] f32 scratch

    const int  nnz  = in_sizes[0];
    const long out4 = (long)out_size >> 2;       // N*128 floats / 4

    long zbl = (out4 + BLOCK - 1) / BLOCK;
    int  zb  = (int)(zbl < 4096 ? zbl : 4096);

    // Init accumulators (d_out/d_ws are poisoned / carry stale data).
    zero_f32_kernel<<<zb, BLOCK, 0, stream>>>((float4*)out, out4);
    zero_hyper_kernel<<<2048, BLOCK, 0, stream>>>((float4*)hyper, dH);

    const int numTiles = (nnz + TILE - 1) / TILE;
    const int grid     = numTiles < 1280 ? numTiles : 1280;  // >1 tile/block
                                                             // -> live prefetch
    // Pass 1: hyper[c] += v * embs[r]
    spmm_scatter_kernel<<<grid, BLOCK, 0, stream>>>(vals, rows, cols, embs, hyper, nnz);
    // Pass 2: out[r] += v * hyper[c]
    spmm_scatter_kernel<<<grid, BLOCK, 0, stream>>>(vals, cols, rows, hyper, out, nnz);
    // LeakyReLU(0.5)
    leaky_kernel<<<zb, BLOCK, 0, stream>>>((float4*)out, out4);
}